// LocalResponseNormalization_70059506532979
// MI455X (gfx1250) — compile-verified
//
#include <hip/hip_runtime.h>
#include <stdint.h>

// LRN over W with channels folded into the window.
// x: [B=32, H=224, W=224, C=64] f32 -> out same shape.
// out = x / sqrt(1 + sum_{w'=w-5..w+5} sum_c x[b,h,w',c]^2)
//
// HBM-bound (822 MB traffic, ~0.4 FLOP/byte -> ~35 us floor at 23.3 TB/s).
//   - one workgroup per (b,h) row (56 KB)
//   - stage row into LDS via GLOBAL_LOAD_ASYNC_TO_LDS_B128 (ASYNCcnt DMA path),
//     nontemporal, immediate offsets carry the 4 KB/chunk global stride
//   - per-w sum of squares + 11-tap window + 1/sqrt entirely in LDS/regs
//   - rescale from LDS, nontemporal coalesced b128 stores

#define BATCH   32
#define HH      224
#define WW      224
#define CC      64
#define RADIUS  5
#define ROW_F   (WW * CC)          // 14336 floats per (b,h) row
#define NT      256                // threads per block (8 wave32)
#define CHUNKS  (ROW_F / 4 / NT)   // 14 float4 chunks per thread
#define STRIDE_F 68                // padded floats per w (64 + 4 pad, 16B aligned)
#define LDS_CHUNK_STEP 256         // (16*STRIDE_F*4) - 4096: LDS-vs-global stride delta

typedef float v4f __attribute__((ext_vector_type(4)));

// One 16B-per-lane async DMA copy, global -> LDS. INST_OFFSET is added to BOTH
// the global VADDR and the LDS VDST address by hardware, so the 4096 B/chunk
// common stride rides in the immediate; only the 256 B padding delta is in the
// LDS VGPR. Input is single-use -> nontemporal.
template<int OFF>
__device__ __forceinline__ void async_cp_b128(uint32_t ldsOff, uint64_t gaddr) {
    asm volatile("global_load_async_to_lds_b128 %0, %1, off offset:%2 th:TH_LOAD_NT"
                 :: "v"(ldsOff), "v"(gaddr), "n"(OFF) : "memory");
}

__global__ __launch_bounds__(NT)
void lrn_w64_kernel(const float* __restrict__ x, float* __restrict__ out) {
    __shared__ float xs[WW * STRIDE_F];  // padded row cache: 60928 B
    __shared__ float s[WW];              // per-w sum of squares
    __shared__ float inv[WW];            // 1/sqrt(1 + window sum)

    const int t  = threadIdx.x;
    const int wt = t >> 4;               // w-slot of chunk 0 (advances by 16/chunk)
    const int jt = t & 15;               // float4 index within a w (chunk-invariant)
    const long long rowBase = (long long)blockIdx.x * ROW_F;
    const float* gx = x + rowBase;

    // ---- Stage: async bulk copy global -> LDS (coalesced 16B per lane) ----
    const uint32_t ldsBase = (uint32_t)(size_t)&xs[0];  // flat low 32 bits == LDS offset
    const uint32_t lds0 = ldsBase + (uint32_t)(wt * (STRIDE_F * 4) + jt * 16);
    const uint64_t g0   = (uint64_t)(const void*)(gx + 4 * t);
#define CP(i) async_cp_b128<(i) * 4096>(lds0 + (i) * LDS_CHUNK_STEP, g0)
    CP(0);  CP(1);  CP(2);  CP(3);  CP(4);  CP(5);  CP(6);
    CP(7);  CP(8);  CP(9);  CP(10); CP(11); CP(12); CP(13);
#undef CP
    asm volatile("s_wait_asynccnt 0x0" ::: "memory");  // this wave's copies done
    __syncthreads();                                   // whole block's copies visible

    // ---- Pass 1: per-w sum of squares (thread t owns w = t) ----
    if (t < WW) {
        const v4f* row = (const v4f*)&xs[t * STRIDE_F];
        float acc = 0.f;
#pragma unroll
        for (int j = 0; j < 16; ++j) {
            v4f v = row[j];
            acc += v.x * v.x + v.y * v.y + v.z * v.z + v.w * v.w;
        }
        s[t] = acc;
    }
    __syncthreads();

    // ---- Pass 2: 11-tap zero-padded window + inverse denominator ----
    if (t < WW) {
        float win = 0.f;
#pragma unroll
        for (int d = -RADIUS; d <= RADIUS; ++d) {
            int idx = t + d;
            if (idx >= 0 && idx < WW) win += s[idx];
        }
        // BIAS=1, ALPHA=1, BETA=0.5  =>  denom = sqrt(1 + win)
        inv[t] = 1.0f / __builtin_sqrtf(1.0f + win);
    }
    __syncthreads();

    // ---- Pass 3: rescale from LDS, nontemporal streaming stores ----
    // Chunk i: w = i*16 + wt, LDS row step = 16*STRIDE_F floats, global step = 1024 floats.
    const float* lrow = &xs[wt * STRIDE_F + jt * 4];
    const float* ivp  = &inv[wt];
    float* gp = out + rowBase + 4 * t;
#pragma unroll
    for (int i = 0; i < CHUNKS; ++i) {
        v4f v   = *(const v4f*)(lrow + i * (16 * STRIDE_F));
        float r = ivp[i * 16];             // 16 lanes same address -> LDS broadcast
        v4f o;
        o.x = v.x * r; o.y = v.y * r; o.z = v.z * r; o.w = v.w * r;
        __builtin_nontemporal_store(o, (v4f*)(gp + i * 1024));
    }
}

extern "C" void kernel_launch(void* const* d_in, const int* in_sizes, int n_in,
                              void* d_out, int out_size, void* d_ws, size_t ws_size,
                              hipStream_t stream) {
    (void)in_sizes; (void)n_in; (void)out_size; (void)d_ws; (void)ws_size;
    const float* x = (const float*)d_in[0];
    float* out = (float*)d_out;
    dim3 grid(BATCH * HH);   // 7168 workgroups, one per (b,h) row
    dim3 block(NT);
    lrn_w64_kernel<<<grid, block, 0, stream>>>(x, out);
}